// SetPooling_86792699118284
// MI455X (gfx1250) — compile-verified
//
#include <hip/hip_runtime.h>

typedef __attribute__((ext_vector_type(2))) float v2f;
typedef __attribute__((ext_vector_type(4))) float v4f;
typedef __attribute__((ext_vector_type(8))) float v8f;

#define TOTAL_ROWS 262144
#define FEAT 512
#define NSEG 1024
#define TILES_PER_WAVE 4   // 8 waves * 4 tiles * 16 cols = 512 features

__global__ __launch_bounds__(256) void seg_mean_wmma_kernel(
    const float* __restrict__ batch,
    const int* __restrict__ seg,
    float* __restrict__ out)
{
    __shared__ float s_mean[FEAT];
    __shared__ int s_bounds[2];

    const int sid = blockIdx.x;      // segment id, one block per segment
    const int tid = threadIdx.x;

    // ---- Phase 0: find contiguous row range [r0, r1) for this segment ----
    if (tid == 0) {
        int lo = 0, hi = TOTAL_ROWS;
        while (lo < hi) { int mid = (lo + hi) >> 1; if (seg[mid] < sid) lo = mid + 1; else hi = mid; }
        s_bounds[0] = lo;
        int lo2 = lo, hi2 = TOTAL_ROWS;
        while (lo2 < hi2) { int mid = (lo2 + hi2) >> 1; if (seg[mid] < sid + 1) lo2 = mid + 1; else hi2 = mid; }
        s_bounds[1] = lo2;
    }
    __syncthreads();

    const int r0 = s_bounds[0];
    const int r1 = s_bounds[1];
    const int count = r1 - r0;

    const int wave = tid >> 5;
    const int lane = tid & 31;
    const int half = lane >> 4;          // 0: lanes 0-15, 1: lanes 16-31
    const int col_in_tile = lane & 15;   // N index within a 16-wide tile

    // Per-lane column base for this wave's 4 tiles (tile t at +t*16).
    const int cbase = wave * (TILES_PER_WAVE * 16) + col_in_tile;
    const float* __restrict__ pbase = batch + cbase;

    // ---- Phase 1: WMMA f32 16x16x4 column-sum reduction ----
    // A = all-ones 16x4  =>  D[m][n] = sum_k B[k][n] + C[m][n] for every m.
    // Invariant to K-slot placement in B, so any fixed row<->slot mapping works.
    v2f ones; ones.x = 1.0f; ones.y = 1.0f;

    v8f acc[TILES_PER_WAVE];
    #pragma unroll
    for (int t = 0; t < TILES_PER_WAVE; ++t) acc[t] = (v8f){};

    int r = r0;
    const int end8 = r0 + (count & ~7);
    const int end4 = r0 + (count & ~3);

    // Main loop: 8 rows per iteration, no guards -> unconditional clustered loads.
    for (; r < end8; r += 8) {
        // half0 handles rows q+0,q+1 ; half1 handles rows q+2,q+3 of each chunk.
        const float* __restrict__ q = pbase + (size_t)(r + half * 2) * FEAT;
        v2f b0[TILES_PER_WAVE], b1[TILES_PER_WAVE];
        #pragma unroll
        for (int t = 0; t < TILES_PER_WAVE; ++t) {
            b0[t].x = q[t * 16];                    // row +0
            b0[t].y = q[FEAT + t * 16];             // row +1
            b1[t].x = q[4 * FEAT + t * 16];         // row +4
            b1[t].y = q[5 * FEAT + t * 16];         // row +5
        }
        #pragma unroll
        for (int t = 0; t < TILES_PER_WAVE; ++t)
            acc[t] = __builtin_amdgcn_wmma_f32_16x16x4_f32(
                false, ones, false, b0[t], (short)0, acc[t], false, false);
        #pragma unroll
        for (int t = 0; t < TILES_PER_WAVE; ++t)
            acc[t] = __builtin_amdgcn_wmma_f32_16x16x4_f32(
                false, ones, false, b1[t], (short)0, acc[t], false, false);
    }

    // One full 4-row chunk, still unguarded.
    if (r < end4) {
        const float* __restrict__ q = pbase + (size_t)(r + half * 2) * FEAT;
        v2f b[TILES_PER_WAVE];
        #pragma unroll
        for (int t = 0; t < TILES_PER_WAVE; ++t) {
            b[t].x = q[t * 16];
            b[t].y = q[FEAT + t * 16];
        }
        #pragma unroll
        for (int t = 0; t < TILES_PER_WAVE; ++t)
            acc[t] = __builtin_amdgcn_wmma_f32_16x16x4_f32(
                false, ones, false, b[t], (short)0, acc[t], false, false);
        r += 4;
    }

    // Guarded tail (<= 3 rows), executed at most once per block.
    if (r < r1) {
        const int rowA = r + half * 2;
        const int rowB = rowA + 1;
        const bool va = rowA < r1;
        const bool vb = rowB < r1;
        const float* __restrict__ qa = pbase + (size_t)rowA * FEAT;
        const float* __restrict__ qb = pbase + (size_t)rowB * FEAT;
        v2f b[TILES_PER_WAVE];
        #pragma unroll
        for (int t = 0; t < TILES_PER_WAVE; ++t) {
            b[t].x = va ? qa[t * 16] : 0.0f;
            b[t].y = vb ? qb[t * 16] : 0.0f;
        }
        #pragma unroll
        for (int t = 0; t < TILES_PER_WAVE; ++t)
            acc[t] = __builtin_amdgcn_wmma_f32_16x16x4_f32(
                false, ones, false, b[t], (short)0, acc[t], false, false);
    }

    // ---- Phase 2: mean = sum / max(count,1) into LDS ----
    const float inv = 1.0f / (float)(count > 0 ? count : 1);
    if (half == 0) {  // lanes 16-31 hold duplicate rows of D; one half suffices
        #pragma unroll
        for (int t = 0; t < TILES_PER_WAVE; ++t)
            s_mean[cbase + t * 16] = acc[t][0] * inv;   // D row M=0 == column sum
    }
    __syncthreads();

    // ---- Phase 3: broadcast mean row to all rows of the segment (float4) ----
    const v4f* __restrict__ m4 = reinterpret_cast<const v4f*>(s_mean);
    v4f* __restrict__ out4 = reinterpret_cast<v4f*>(out);
    const int Q = FEAT / 4;  // 128 float4 per row
    const int total4 = count * Q;
    for (int i = tid; i < total4; i += 256) {
        const int row = r0 + (i >> 7);
        const int c = i & (Q - 1);
        out4[(size_t)row * Q + c] = m4[c];
    }
}

extern "C" void kernel_launch(void* const* d_in, const int* in_sizes, int n_in,
                              void* d_out, int out_size, void* d_ws, size_t ws_size,
                              hipStream_t stream) {
    const float* batch = (const float*)d_in[0];
    const int* seg     = (const int*)d_in[1];
    // d_in[2] = num_segments scalar (fixed at 1024 for this problem shape)
    float* out = (float*)d_out;

    seg_mean_wmma_kernel<<<NSEG, 256, 0, stream>>>(batch, seg, out);
}